// E_GCL_76416058130599
// MI455X (gfx1250) — compile-verified
//
#include <hip/hip_runtime.h>
#include <hip/hip_bf16.h>

typedef __attribute__((ext_vector_type(2))) float v2f;
typedef __attribute__((ext_vector_type(8))) float v8f;

#define WAVES_PER_BLOCK 8

__device__ inline v8f wmma_f32_k4(v2f a, v2f b, v8f c) {
    // D = A(16x4 f32) * B(4x16 f32) + C(16x16 f32)
    return __builtin_amdgcn_wmma_f32_16x16x4_f32(
        /*neg_a=*/false, a, /*neg_b=*/false, b,
        /*c_mod=*/(short)0, c, /*reuse_a=*/false, /*reuse_b=*/false);
}

__device__ inline void atomic_add_f32(float* p, float v) {
    __hip_atomic_fetch_add(p, v, __ATOMIC_RELAXED, __HIP_MEMORY_SCOPE_AGENT);
}

// ---------------------------------------------------------------------------
// Edge kernel: per 16-edge tile, edge_in = [h[row] | h[col] | radial] (K=129,
// padded to 132), MLP 129->64 (ReLU) ->64 (ReLU) * edge_mask, then
// atomic scatter-add into agg[row].
// ---------------------------------------------------------------------------
__launch_bounds__(256)
__global__ void egcl_edge_kernel(const float* __restrict__ h,
                                 const float* __restrict__ coord,
                                 const int*   __restrict__ row,
                                 const int*   __restrict__ col,
                                 const float* __restrict__ edge_mask,
                                 const float* __restrict__ W_e1,
                                 const float* __restrict__ b_e1,
                                 const float* __restrict__ W_e2,
                                 const float* __restrict__ b_e2,
                                 float* __restrict__ agg,
                                 int E)
{
    __shared__ v2f   W1p[66 * 64];                     // paired rows of W_e1 (K padded 129->132)
    __shared__ v2f   W2p[32 * 64];                     // paired rows of W_e2 (K=64)
    __shared__ float bias1[64];
    __shared__ float bias2[64];
    __shared__ float stage[WAVES_PER_BLOCK][16 * 64];  // layer-1 activations per wave
    __shared__ int   rowsL[WAVES_PER_BLOCK][16];
    __shared__ float maskL[WAVES_PER_BLOCK][16];

    const int tid = threadIdx.x;

    for (int i = tid; i < 66 * 64; i += blockDim.x) {
        int kp = i >> 6, n = i & 63;
        float a = (2 * kp     < 129) ? W_e1[(2 * kp    ) * 64 + n] : 0.f;
        float b = (2 * kp + 1 < 129) ? W_e1[(2 * kp + 1) * 64 + n] : 0.f;
        v2f w = {a, b};
        W1p[i] = w;
    }
    for (int i = tid; i < 32 * 64; i += blockDim.x) {
        int kp = i >> 6, n = i & 63;
        v2f w = {W_e2[(2 * kp) * 64 + n], W_e2[(2 * kp + 1) * 64 + n]};
        W2p[i] = w;
    }
    if (tid < 64) { bias1[tid] = b_e1[tid]; bias2[tid] = b_e2[tid]; }
    __syncthreads();

    const int lane = tid & 31;
    const int wid  = tid >> 5;
    const int m    = lane & 15;   // tile-row (edge) / B,C column index
    const int kh   = lane >> 4;   // K-half select (0 or 1)
    const int ntiles = (E + 15) >> 4;
    const int wave_global = blockIdx.x * WAVES_PER_BLOCK + wid;
    const int nwaves = gridDim.x * WAVES_PER_BLOCK;

    float* st = stage[wid];
    int*   rl = rowsL[wid];
    float* ml = maskL[wid];

    for (int tile = wave_global; tile < ntiles; tile += nwaves) {
        const int e_raw = (tile << 4) + m;
        const bool valid = e_raw < E;
        const int e = valid ? e_raw : (E - 1);
        const int r = row[e];
        const int c = col[e];
        if (lane < 16) {
            rl[m] = r;
            ml[m] = valid ? edge_mask[e] : 0.f;
        }
        float dx = coord[r * 3 + 0] - coord[c * 3 + 0];
        float dy = coord[r * 3 + 1] - coord[c * 3 + 1];
        float dz = coord[r * 3 + 2] - coord[c * 3 + 2];
        float radial = dx * dx + dy * dy + dz * dz;

        const float* hr = h + (size_t)r * 64;
        const float* hc = h + (size_t)c * 64;

        v8f acc0 = {}, acc1 = {}, acc2 = {}, acc3 = {};

        // ---- layer 1: K = 132 (129 real + 3 zero pad) ----
        #pragma unroll
        for (int kb = 0; kb < 132; kb += 4) {
            v2f a;
            if (kb < 64) {
                a = *(const v2f*)(hr + kb + 2 * kh);
            } else if (kb < 128) {
                a = *(const v2f*)(hc + (kb - 64) + 2 * kh);
            } else {
                v2f ar = {radial, 0.f};
                v2f az = {0.f, 0.f};
                a = (kh == 0) ? ar : az;
            }
            const int kp = (kb >> 1) + kh;
            acc0 = wmma_f32_k4(a, W1p[kp * 64 +  0 + m], acc0);
            acc1 = wmma_f32_k4(a, W1p[kp * 64 + 16 + m], acc1);
            acc2 = wmma_f32_k4(a, W1p[kp * 64 + 32 + m], acc2);
            acc3 = wmma_f32_k4(a, W1p[kp * 64 + 48 + m], acc3);
        }

        // bias + ReLU, restage C-layout -> row-major LDS tile
        {
            v8f t0 = acc0, t1 = acc1, t2 = acc2, t3 = acc3;
            #pragma unroll
            for (int t = 0; t < 4; ++t) {
                const v8f cv = (t == 0) ? t0 : (t == 1) ? t1 : (t == 2) ? t2 : t3;
                const float bv = bias1[16 * t + m];
                #pragma unroll
                for (int rr = 0; rr < 8; ++rr) {
                    float v = cv[rr] + bv;
                    v = v > 0.f ? v : 0.f;
                    st[(rr + 8 * kh) * 64 + 16 * t + m] = v;
                }
            }
        }

        // ---- layer 2: K = 64 ----
        v8f o0 = {}, o1 = {}, o2 = {}, o3 = {};
        #pragma unroll
        for (int kb = 0; kb < 64; kb += 4) {
            v2f a = *(const v2f*)(st + m * 64 + kb + 2 * kh);
            const int kp = (kb >> 1) + kh;
            o0 = wmma_f32_k4(a, W2p[kp * 64 +  0 + m], o0);
            o1 = wmma_f32_k4(a, W2p[kp * 64 + 16 + m], o1);
            o2 = wmma_f32_k4(a, W2p[kp * 64 + 32 + m], o2);
            o3 = wmma_f32_k4(a, W2p[kp * 64 + 48 + m], o3);
        }

        // bias + ReLU + edge_mask, atomic scatter into agg[row]
        #pragma unroll
        for (int t = 0; t < 4; ++t) {
            const v8f cv = (t == 0) ? o0 : (t == 1) ? o1 : (t == 2) ? o2 : o3;
            const float bv = bias2[16 * t + m];
            #pragma unroll
            for (int rr = 0; rr < 8; ++rr) {
                const int mi = rr + 8 * kh;
                float v = cv[rr] + bv;
                v = v > 0.f ? v : 0.f;
                v *= ml[mi];
                const int rn = rl[mi];
                atomic_add_f32(&agg[(size_t)rn * 64 + 16 * t + m], v);
            }
        }
    }
}

// ---------------------------------------------------------------------------
// Node kernel: node_in = [h | agg] (K=128), MLP 128->64 (ReLU) ->64, residual
// h_out = h + out.
// ---------------------------------------------------------------------------
__launch_bounds__(256)
__global__ void egcl_node_kernel(const float* __restrict__ h,
                                 const float* __restrict__ agg,
                                 const float* __restrict__ W_n1,
                                 const float* __restrict__ b_n1,
                                 const float* __restrict__ W_n2,
                                 const float* __restrict__ b_n2,
                                 float* __restrict__ h_out,
                                 int N)
{
    __shared__ v2f   W1p[64 * 64];   // paired rows of W_n1 (K=128)
    __shared__ v2f   W2p[32 * 64];   // paired rows of W_n2 (K=64)
    __shared__ float bias1[64];
    __shared__ float bias2[64];
    __shared__ float stage[WAVES_PER_BLOCK][16 * 64];

    const int tid = threadIdx.x;
    for (int i = tid; i < 64 * 64; i += blockDim.x) {
        int kp = i >> 6, n = i & 63;
        v2f w = {W_n1[(2 * kp) * 64 + n], W_n1[(2 * kp + 1) * 64 + n]};
        W1p[i] = w;
    }
    for (int i = tid; i < 32 * 64; i += blockDim.x) {
        int kp = i >> 6, n = i & 63;
        v2f w = {W_n2[(2 * kp) * 64 + n], W_n2[(2 * kp + 1) * 64 + n]};
        W2p[i] = w;
    }
    if (tid < 64) { bias1[tid] = b_n1[tid]; bias2[tid] = b_n2[tid]; }
    __syncthreads();

    const int lane = tid & 31;
    const int wid  = tid >> 5;
    const int m    = lane & 15;
    const int kh   = lane >> 4;
    const int ntiles = (N + 15) >> 4;
    const int wave_global = blockIdx.x * WAVES_PER_BLOCK + wid;
    const int nwaves = gridDim.x * WAVES_PER_BLOCK;
    float* st = stage[wid];

    for (int tile = wave_global; tile < ntiles; tile += nwaves) {
        const int node_raw = (tile << 4) + m;
        const int node = node_raw < N ? node_raw : (N - 1);
        const float* hrow = h   + (size_t)node * 64;
        const float* arow = agg + (size_t)node * 64;

        v8f acc0 = {}, acc1 = {}, acc2 = {}, acc3 = {};
        #pragma unroll
        for (int kb = 0; kb < 128; kb += 4) {
            v2f a;
            if (kb < 64) a = *(const v2f*)(hrow + kb + 2 * kh);
            else         a = *(const v2f*)(arow + (kb - 64) + 2 * kh);
            const int kp = (kb >> 1) + kh;
            acc0 = wmma_f32_k4(a, W1p[kp * 64 +  0 + m], acc0);
            acc1 = wmma_f32_k4(a, W1p[kp * 64 + 16 + m], acc1);
            acc2 = wmma_f32_k4(a, W1p[kp * 64 + 32 + m], acc2);
            acc3 = wmma_f32_k4(a, W1p[kp * 64 + 48 + m], acc3);
        }

        #pragma unroll
        for (int t = 0; t < 4; ++t) {
            const v8f cv = (t == 0) ? acc0 : (t == 1) ? acc1 : (t == 2) ? acc2 : acc3;
            const float bv = bias1[16 * t + m];
            #pragma unroll
            for (int rr = 0; rr < 8; ++rr) {
                float v = cv[rr] + bv;
                v = v > 0.f ? v : 0.f;
                st[(rr + 8 * kh) * 64 + 16 * t + m] = v;
            }
        }

        v8f o0 = {}, o1 = {}, o2 = {}, o3 = {};
        #pragma unroll
        for (int kb = 0; kb < 64; kb += 4) {
            v2f a = *(const v2f*)(st + m * 64 + kb + 2 * kh);
            const int kp = (kb >> 1) + kh;
            o0 = wmma_f32_k4(a, W2p[kp * 64 +  0 + m], o0);
            o1 = wmma_f32_k4(a, W2p[kp * 64 + 16 + m], o1);
            o2 = wmma_f32_k4(a, W2p[kp * 64 + 32 + m], o2);
            o3 = wmma_f32_k4(a, W2p[kp * 64 + 48 + m], o3);
        }

        // bias (no ReLU) + residual, write h_out
        #pragma unroll
        for (int t = 0; t < 4; ++t) {
            const v8f cv = (t == 0) ? o0 : (t == 1) ? o1 : (t == 2) ? o2 : o3;
            const float bv = bias2[16 * t + m];
            #pragma unroll
            for (int rr = 0; rr < 8; ++rr) {
                const int mi = rr + 8 * kh;
                const int nd = (tile << 4) + mi;
                if (nd < N) {
                    const int cn = 16 * t + m;
                    float v = cv[rr] + bv + h[(size_t)nd * 64 + cn];
                    h_out[(size_t)nd * 64 + cn] = v;
                }
            }
        }
    }
}

__global__ void zero_f32_kernel(float* __restrict__ p, int n) {
    int i = blockIdx.x * blockDim.x + threadIdx.x;
    if (i < n) p[i] = 0.f;
}

__global__ void copy_f32_kernel(const float* __restrict__ src, float* __restrict__ dst, int n) {
    int i = blockIdx.x * blockDim.x + threadIdx.x;
    if (i < n) dst[i] = src[i];
}

extern "C" void kernel_launch(void* const* d_in, const int* in_sizes, int n_in,
                              void* d_out, int out_size, void* d_ws, size_t ws_size,
                              hipStream_t stream) {
    const float* h         = (const float*)d_in[0];
    const float* coord     = (const float*)d_in[1];
    const int*   row       = (const int*)  d_in[2];
    const int*   col       = (const int*)  d_in[3];
    // d_in[4] = node_mask (unused by the reference forward path)
    const float* edge_mask = (const float*)d_in[5];
    const float* W_e1      = (const float*)d_in[6];
    const float* b_e1      = (const float*)d_in[7];
    const float* W_e2      = (const float*)d_in[8];
    const float* b_e2      = (const float*)d_in[9];
    const float* W_n1      = (const float*)d_in[10];
    const float* b_n1      = (const float*)d_in[11];
    const float* W_n2      = (const float*)d_in[12];
    const float* b_n2      = (const float*)d_in[13];

    const int N = in_sizes[0] / 64;   // h is [N, 64]
    const int E = in_sizes[2];        // row is [E]

    float* agg       = (float*)d_ws;          // [N, 64] scratch
    float* h_out     = (float*)d_out;         // [N, 64]
    float* coord_out = h_out + (size_t)N * 64; // [N, 3]

    const int aggN = N * 64;
    zero_f32_kernel<<<(aggN + 255) / 256, 256, 0, stream>>>(agg, aggN);

    egcl_edge_kernel<<<1250, 256, 0, stream>>>(h, coord, row, col, edge_mask,
                                               W_e1, b_e1, W_e2, b_e2, agg, E);

    egcl_node_kernel<<<200, 256, 0, stream>>>(h, agg, W_n1, b_n1, W_n2, b_n2,
                                              h_out, N);

    const int nc = N * 3;
    copy_f32_kernel<<<(nc + 255) / 256, 256, 0, stream>>>(coord, coord_out, nc);
}